// RW_NN_27685359190239
// MI455X (gfx1250) — compile-verified
//
#include <hip/hip_runtime.h>
#include <math.h>

// ---------------- problem constants ----------------
constexpr int NG  = 256;          // graphs
constexpr int NPG = 256;          // nodes per graph
constexpr int NTOT = NG * NPG;    // 65536
constexpr int EDGES = 1048576;
constexpr int IN_DIM = 128;
constexpr int HD = 64;
constexpr int HG = 16;
constexpr int SHG = 6;
constexpr int MS = 3;
constexpr int PEN = 256;
constexpr int NCLS = 10;

typedef __attribute__((ext_vector_type(2))) float v2f;
typedef __attribute__((ext_vector_type(8))) float v8f;

// ---------------- small helpers ----------------
__device__ __forceinline__ float waveReduceSum(float v) {
  #pragma unroll
  for (int off = 16; off > 0; off >>= 1) v += __shfl_down(v, off, 32);
  return v;
}

__global__ void zero_kernel(float* p, long long n) {
  long long i = (long long)blockIdx.x * blockDim.x + threadIdx.x;
  long long stride = (long long)gridDim.x * blockDim.x;
  for (; i < n; i += stride) p[i] = 0.0f;
}

// deg[src] += w  (segment_sum of edge_weight by edge_src)
__global__ void deg_kernel(const int* __restrict__ src, const float* __restrict__ w,
                           float* __restrict__ deg) {
  int e = blockIdx.x * blockDim.x + threadIdx.x;
  if (e < EDGES) atomicAdd(&deg[src[e]], w[e]);
}

// norm_div[g] = NPG - #(deg==0 in graph g)
__global__ void norm_kernel(const float* __restrict__ deg, float* __restrict__ norm_div) {
  int g = blockIdx.x, t = threadIdx.x;
  int z = (deg[g * NPG + t] == 0.0f) ? 1 : 0;
  __shared__ int sh[8];
  float s = waveReduceSum((float)z);
  if ((t & 31) == 0) sh[t >> 5] = (int)s;
  __syncthreads();
  if (t == 0) {
    int S = 0;
    #pragma unroll
    for (int w = 0; w < 8; w++) S += sh[w];
    norm_div[g] = (float)(NPG - S);
  }
}

// ---------------- x0 = sigmoid(features @ fc_w^T), WMMA f32 16x16x4 ----------------
// one wave -> one 16x16 tile of [NTOT x HD]; K = IN_DIM = 128 -> 32 WMMA steps
__global__ void x0_gemm_kernel(const float* __restrict__ features,
                               const float* __restrict__ fc_w,
                               float* __restrict__ x0) {
  int waveg = blockIdx.x * (blockDim.x >> 5) + (threadIdx.x >> 5);
  int lane  = threadIdx.x & 31;
  int tm = waveg >> 2;            // row tile 0..4095
  int tn = waveg & 3;             // col tile 0..3
  int half = lane >> 4, lml = lane & 15;
  int m = tm * 16 + lml;          // A row owned by this lane
  int n = tn * 16 + lml;          // B col owned by this lane
  v8f c = {};
  for (int k0 = 0; k0 < IN_DIM; k0 += 4) {
    int ka = k0 + 2 * half;
    v2f a, b;
    a.x = features[(long long)m * IN_DIM + ka];
    a.y = features[(long long)m * IN_DIM + ka + 1];
    // B[k,n] = fc_w[n,k]
    b.x = fc_w[(long long)n * IN_DIM + ka];
    b.y = fc_w[(long long)n * IN_DIM + ka + 1];
    c = __builtin_amdgcn_wmma_f32_16x16x4_f32(false, a, false, b, (short)0, c, false, false);
  }
  int row0 = tm * 16 + 8 * half;
  int col  = tn * 16 + lml;
  #pragma unroll
  for (int r = 0; r < 8; r++) {
    float v = c[r];
    x0[(long long)(row0 + r) * HD + col] = 1.0f / (1.0f + expf(-v));
  }
}

// ---------------- T_i[d] = t[d]^T @ x_i[d]  (64x64, K=256), WMMA f32 ----------------
// grid: NG*2 blocks of 8 waves; each wave does one of 16 tiles of the 64x64 output
__global__ void ti_gemm_kernel(const float* __restrict__ tmat,   // x0
                               const float* __restrict__ xi,     // x_i
                               const float* __restrict__ deg,
                               float* __restrict__ Ti, int masked) {
  int bid = blockIdx.x;
  int d   = bid >> 1;
  int wv  = (bid & 1) * 8 + (threadIdx.x >> 5);   // tile 0..15
  int tm = wv >> 2, tn = wv & 3;
  int lane = threadIdx.x & 31;
  int half = lane >> 4, lml = lane & 15;
  const float* tg = tmat + (long long)d * NPG * HD;
  const float* xg = xi   + (long long)d * NPG * HD;
  const float* dg = deg  + (long long)d * NPG;
  int ca = tm * 16 + lml;   // A row = c index
  int cb = tn * 16 + lml;   // B col = c' index
  v8f acc = {};
  for (int k0 = 0; k0 < NPG; k0 += 4) {
    int e0 = k0 + 2 * half;
    int e1 = e0 + 1;
    v2f a, b;
    a.x = tg[e0 * HD + ca];           // A[c,e] = t[e,c]
    a.y = tg[e1 * HD + ca];
    float m0 = 1.0f, m1 = 1.0f;
    if (masked) {                      // step 0: B uses adj0 * t
      m0 = dg[e0] > 0.0f ? 1.0f : 0.0f;
      m1 = dg[e1] > 0.0f ? 1.0f : 0.0f;
    }
    b.x = xg[e0 * HD + cb] * m0;       // B[e,c'] = x_i[e,c']
    b.y = xg[e1 * HD + cb] * m1;
    acc = __builtin_amdgcn_wmma_f32_16x16x4_f32(false, a, false, b, (short)0, acc, false, false);
  }
  int r0 = tm * 16 + 8 * half;
  #pragma unroll
  for (int r = 0; r < 8; r++)
    Ti[(long long)d * 4096 + (r0 + r) * HD + (tn * 16 + lml)] = acc[r];
}

// ---------------- SpMM: y[src] += w * x[dst], 64-wide rows ----------------
__global__ void spmm_kernel(const int* __restrict__ src, const int* __restrict__ dst,
                            const float* __restrict__ w, const float* __restrict__ x,
                            float* __restrict__ y) {
  long long gid = (long long)blockIdx.x * blockDim.x + threadIdx.x;
  int e = (int)(gid >> 6);
  int j = (int)(gid & 63);
  if (e < EDGES) {
    float v = w[e] * x[(long long)dst[e] * HD + j];
    atomicAdd(&y[(long long)src[e] * HD + j], v);
  }
}

// ---------------- Z_i[a] = z0[a]^T z_i[a] with z_i = A^i z0 (tiny) ----------------
__constant__ int IU_R[15] = {0,0,0,0,0,1,1,1,1,2,2,2,3,3,4};
__constant__ int IU_C[15] = {1,2,3,4,5,2,3,4,5,3,4,5,4,5,5};

__global__ void z_kernel(const float* __restrict__ adj_hidden,
                         const float* __restrict__ features_hidden,
                         float* __restrict__ Zb) {
  int a = blockIdx.x, t = threadIdx.x;
  __shared__ float A6[6][6];
  __shared__ float z0[SHG * HD], z1[SHG * HD], z2[SHG * HD];
  if (t < 36) A6[t / 6][t % 6] = 0.0f;
  __syncthreads();
  if (t < 15) {
    float v = adj_hidden[a * 15 + t];
    v = v > 0.0f ? v : 0.0f;                 // relu
    A6[IU_R[t]][IU_C[t]] = v;
    A6[IU_C[t]][IU_R[t]] = v;                // symmetric
  }
  for (int idx = t; idx < SHG * HD; idx += blockDim.x)
    z0[idx] = features_hidden[a * SHG * HD + idx];
  __syncthreads();
  for (int idx = t; idx < SHG * HD; idx += blockDim.x) {
    int b = idx / HD, c = idx % HD;
    float s = 0.0f;
    #pragma unroll
    for (int bb = 0; bb < SHG; bb++) s += A6[b][bb] * z0[bb * HD + c];
    z1[idx] = s;
  }
  __syncthreads();
  for (int idx = t; idx < SHG * HD; idx += blockDim.x) {
    int b = idx / HD, c = idx % HD;
    float s = 0.0f;
    #pragma unroll
    for (int bb = 0; bb < SHG; bb++) s += A6[b][bb] * z1[bb * HD + c];
    z2[idx] = s;
  }
  __syncthreads();
  for (int idx = t; idx < HD * HD; idx += blockDim.x) {
    int c = idx / HD, cp = idx % HD;
    float s0 = 0.0f, s1 = 0.0f, s2 = 0.0f;
    #pragma unroll
    for (int b = 0; b < SHG; b++) {
      float zc = z0[b * HD + c];
      s0 += zc * z0[b * HD + cp];
      s1 += zc * z1[b * HD + cp];
      s2 += zc * z2[b * HD + cp];
    }
    Zb[(0 * HG + a) * 4096 + idx] = s0;
    Zb[(1 * HG + a) * 4096 + idx] = s1;
    Zb[(2 * HG + a) * 4096 + idx] = s2;
  }
}

// ---------------- s_i[d,a] = <Z_i[a], T_i[d]> / norm_div[d] ----------------
// grid: NG*6 blocks of 8 waves; wave -> one (i,a) pair; pair = i*16+a = output col
__global__ void s_kernel(const float* __restrict__ Zb, const float* __restrict__ Ti,
                         const float* __restrict__ norm_div, float* __restrict__ out_pre) {
  int bid = blockIdx.x;
  int d = bid / 6, grp = bid % 6;
  int wave = threadIdx.x >> 5, lane = threadIdx.x & 31;
  int pair = grp * 8 + wave;            // 0..47
  int i = pair / HG, a = pair % HG;
  const float* Zp = Zb + (long long)(i * HG + a) * 4096;
  const float* Tp = Ti + (long long)(i * NG + d) * 4096;
  float s = 0.0f;
  for (int k = lane; k < 4096; k += 32) s += Zp[k] * Tp[k];
  s = waveReduceSum(s);
  if (lane == 0) out_pre[d * (HG * MS) + pair] = s / norm_div[d];
}

// ---------------- BatchNorm batch statistics over 256 rows ----------------
__global__ void bn_stats_kernel(const float* __restrict__ out_pre,
                                float* __restrict__ mean, float* __restrict__ rstd) {
  int f = blockIdx.x, t = threadIdx.x;   // 48 blocks x 256 threads
  float v = out_pre[t * (HG * MS) + f];
  __shared__ float ssum[8], ssq[8];
  float s = waveReduceSum(v);
  float q = waveReduceSum(v * v);
  if ((t & 31) == 0) { ssum[t >> 5] = s; ssq[t >> 5] = q; }
  __syncthreads();
  if (t == 0) {
    float S = 0.0f, Q = 0.0f;
    #pragma unroll
    for (int w = 0; w < 8; w++) { S += ssum[w]; Q += ssq[w]; }
    float m = S / (float)NG;
    float var = Q / (float)NG - m * m;
    mean[f] = m;
    rstd[f] = rsqrtf(var + 1e-5f);
  }
}

// ---------------- head: BN-apply -> fc1+relu -> fc2 -> log_softmax ----------------
__global__ void head_kernel(const float* __restrict__ out_pre,
                            const float* __restrict__ mean, const float* __restrict__ rstd,
                            const float* __restrict__ gamma, const float* __restrict__ beta,
                            const float* __restrict__ fc1_w, const float* __restrict__ fc1_b,
                            const float* __restrict__ fc2_w, const float* __restrict__ fc2_b,
                            float* __restrict__ out) {
  int d = blockIdx.x, t = threadIdx.x;    // 256 blocks x 256 threads
  __shared__ float xn[HG * MS];
  __shared__ float h[PEN];
  __shared__ float logits[NCLS];
  if (t < HG * MS)
    xn[t] = gamma[t] * (out_pre[d * (HG * MS) + t] - mean[t]) * rstd[t] + beta[t];
  __syncthreads();
  float acc = fc1_b[t];
  #pragma unroll 8
  for (int f = 0; f < HG * MS; f++) acc += xn[f] * fc1_w[t * (HG * MS) + f];
  h[t] = fmaxf(acc, 0.0f);
  __syncthreads();
  if (t < NCLS) {
    float l = fc2_b[t];
    for (int p = 0; p < PEN; p++) l += h[p] * fc2_w[t * PEN + p];
    logits[t] = l;
  }
  __syncthreads();
  if (t == 0) {
    float mx = logits[0];
    #pragma unroll
    for (int c = 1; c < NCLS; c++) mx = fmaxf(mx, logits[c]);
    float se = 0.0f;
    #pragma unroll
    for (int c = 0; c < NCLS; c++) se += expf(logits[c] - mx);
    float lse = mx + logf(se);
    #pragma unroll
    for (int c = 0; c < NCLS; c++) out[d * NCLS + c] = logits[c] - lse;
  }
}

// ---------------- launcher ----------------
extern "C" void kernel_launch(void* const* d_in, const int* in_sizes, int n_in,
                              void* d_out, int out_size, void* d_ws, size_t ws_size,
                              hipStream_t stream) {
  const float* features        = (const float*)d_in[0];
  const float* edge_weight     = (const float*)d_in[1];
  const float* adj_hidden      = (const float*)d_in[2];
  const float* features_hidden = (const float*)d_in[3];
  const float* fc_w            = (const float*)d_in[4];
  const float* bn_gamma        = (const float*)d_in[5];
  const float* bn_beta         = (const float*)d_in[6];
  const float* fc1_w           = (const float*)d_in[7];
  const float* fc1_b           = (const float*)d_in[8];
  const float* fc2_w           = (const float*)d_in[9];
  const float* fc2_b           = (const float*)d_in[10];
  const int*   edge_src        = (const int*)d_in[11];
  const int*   edge_dst        = (const int*)d_in[12];
  float* out = (float*)d_out;

  float* ws = (float*)d_ws;
  size_t off = 0;
  float* deg      = ws + off; off += NTOT;
  float* x0       = ws + off; off += (size_t)NTOT * HD;
  float* x1       = ws + off; off += (size_t)NTOT * HD;
  float* x2       = ws + off; off += (size_t)NTOT * HD;
  float* Ti       = ws + off; off += (size_t)MS * NG * 4096;   // [3][256][64x64]
  float* Zb       = ws + off; off += (size_t)MS * HG * 4096;   // [3][16][64x64]
  float* out_pre  = ws + off; off += (size_t)NG * HG * MS;
  float* norm_div = ws + off; off += NG;
  float* bn_mean  = ws + off; off += HG * MS;
  float* bn_rstd  = ws + off; off += HG * MS;

  // 1) degree + per-graph normalization
  zero_kernel<<<256, 256, 0, stream>>>(deg, (long long)NTOT);
  deg_kernel<<<(EDGES + 255) / 256, 256, 0, stream>>>(edge_src, edge_weight, deg);
  norm_kernel<<<NG, NPG, 0, stream>>>(deg, norm_div);

  // 2) x0 = sigmoid(features @ fc_w^T)  — WMMA
  {
    int tiles = (NTOT / 16) * (HD / 16);          // 16384 waves
    x0_gemm_kernel<<<tiles / 8, 256, 0, stream>>>(features, fc_w, x0);
  }

  // 3) T_0 (masked), then SpMM hops interleaved with T_1, T_2 — WMMA
  ti_gemm_kernel<<<NG * 2, 256, 0, stream>>>(x0, x0, deg, Ti + 0 * (size_t)NG * 4096, 1);

  zero_kernel<<<2048, 256, 0, stream>>>(x1, (long long)NTOT * HD);
  spmm_kernel<<<(int)(((long long)EDGES * 64) / 256), 256, 0, stream>>>(edge_src, edge_dst,
                                                                        edge_weight, x0, x1);
  ti_gemm_kernel<<<NG * 2, 256, 0, stream>>>(x0, x1, deg, Ti + 1 * (size_t)NG * 4096, 0);

  zero_kernel<<<2048, 256, 0, stream>>>(x2, (long long)NTOT * HD);
  spmm_kernel<<<(int)(((long long)EDGES * 64) / 256), 256, 0, stream>>>(edge_src, edge_dst,
                                                                        edge_weight, x1, x2);
  ti_gemm_kernel<<<NG * 2, 256, 0, stream>>>(x0, x2, deg, Ti + 2 * (size_t)NG * 4096, 0);

  // 4) hidden-graph Gram tensors Z_i
  z_kernel<<<HG, 256, 0, stream>>>(adj_hidden, features_hidden, Zb);

  // 5) scores, batch-norm stats, head
  s_kernel<<<NG * 6, 256, 0, stream>>>(Zb, Ti, norm_div, out_pre);
  bn_stats_kernel<<<HG * MS, NG, 0, stream>>>(out_pre, bn_mean, bn_rstd);
  head_kernel<<<NG, PEN, 0, stream>>>(out_pre, bn_mean, bn_rstd, bn_gamma, bn_beta,
                                      fc1_w, fc1_b, fc2_w, fc2_b, out);
}